// GroupAvgPool1d_47931835023909
// MI455X (gfx1250) — compile-verified
//
#include <hip/hip_runtime.h>

// GroupAvgPool1d: feature[b,g,c] = mean_n( [y[b,n]==g] * x[b,n,c] ), mask[b,g]=count>0
// B=16, N=8192, C=64, G=512.  x: f32 [B,N,C], y: i32 [B,N] (JAX x64 off).
// d_out = [B*G*C feature f32][B*G mask f32], concatenated flat.
//
// Memory-bound scatter-reduce (~35 MB @ 23.3 TB/s). CDNA5 leverage:
//  * 320 KB LDS/WGP -> full privatized [512 x 64] f32 accumulator per chunk
//    (139 KB with bank-conflict padding; impossible on CDNA4's 64 KB).
//  * global_load_async_to_lds_b128 (ASYNCcnt) to stage group ids.
//  * ds_add_f32 scatter with a 68-float padded row stride so the two 16-lane
//    point slices of a wave32 land on disjoint LDS banks (stride 64 would
//    guarantee a 2-way conflict on every atomic).

namespace {
constexpr int kB = 16;
constexpr int kN = 8192;
constexpr int kC = 64;
constexpr int kG = 512;
constexpr int kChunks = 16;              // chunks per batch -> 256 workgroups
constexpr int kPts = kN / kChunks;       // 512 points per workgroup
constexpr int kThreads = 512;            // 16 wave32 per workgroup
constexpr int kAccStride = kC + 4;       // 68 floats: rotates banks by 4*g mod 64
constexpr int kFeatElems = kB * kG * kC; // 524288
constexpr int kMaskElems = kB * kG;      // 8192
constexpr int kAccF4 = kG * kAccStride / 4;  // 8704 float4 to zero
}

__global__ __launch_bounds__(256) void zero_out_kernel(float4* __restrict__ out4, int n4) {
  int i = blockIdx.x * blockDim.x + threadIdx.x;
  if (i < n4) out4[i] = make_float4(0.f, 0.f, 0.f, 0.f);
}

__global__ __launch_bounds__(kThreads) void scatter_kernel(const float* __restrict__ x,
                                                           const int* __restrict__ y,
                                                           float* __restrict__ out) {
  extern __shared__ char smem[];
  float*    acc = (float*)smem;                                   // kG*kAccStride f32 (136 KB)
  unsigned* cnt = (unsigned*)(smem + (size_t)kG * kAccStride * sizeof(float)); // kG u32
  int*      ys  = (int*)((char*)cnt + kG * sizeof(unsigned));     // kPts i32

  const int tid = threadIdx.x;
  const int b   = blockIdx.y;
  const int n0  = blockIdx.x * kPts;

  // ---- stage this chunk's group ids into LDS via CDNA5 async DMA (ASYNCcnt) ----
  const int* yg = y + (size_t)b * kN + n0;
  if (tid < kPts / 4) {   // 128 lanes x b128 = 512 ints
    unsigned   l = (unsigned)(unsigned long long)(ys + tid * 4); // low 32b of generic LDS ptr = LDS offset
    const int* g = yg + tid * 4;
    asm volatile("global_load_async_to_lds_b128 %0, %1, off" :: "v"(l), "v"(g) : "memory");
  }

  // ---- zero accumulator + counts while the async DMA is in flight ----
  float4* a4 = (float4*)acc;
  for (int i = tid; i < kAccF4; i += kThreads)
    a4[i] = make_float4(0.f, 0.f, 0.f, 0.f);
  cnt[tid & (kG - 1)] = 0u;   // kThreads==512==kG: each slot written once

  asm volatile("s_wait_asynccnt 0x0" ::: "memory");
  __syncthreads();

  // ---- scatter: 16 lanes per point, each lane owns 4 consecutive channels ----
  const int cq = tid & 15;   // channel quad 0..15
  const int pr = tid >> 4;   // point row    0..31
  const float4* x4 = (const float4*)x;
  const size_t xbase = ((size_t)b * kN + n0) * (kC / 4); // in float4 units

  #pragma unroll 4
  for (int it = 0; it < kPts / 32; ++it) {               // 16 iterations
    const int p = it * 32 + pr;
    const int g = ys[p];
    const float4 v = x4[xbase + (size_t)p * (kC / 4) + cq];   // global_load_b128
    if ((unsigned)g < (unsigned)kG) {                    // y<0 -> dropped (sentinel class)
      const int base = g * kAccStride + cq * 4;
      atomicAdd(&acc[base + 0], v.x);                    // ds_add_f32
      atomicAdd(&acc[base + 1], v.y);
      atomicAdd(&acc[base + 2], v.z);
      atomicAdd(&acc[base + 3], v.w);
      if (cq == 0) atomicAdd(&cnt[g], 1u);               // ds_add_u32
    }
  }
  __syncthreads();

  // ---- flush: only touched groups go to global atomics ----
  float* facc = out + (size_t)b * kG * kC;
  for (int i = tid; i < kG * kC; i += kThreads) {
    const int g = i >> 6, c = i & (kC - 1);
    if (cnt[g] != 0u) atomicAdd(&facc[i], acc[g * kAccStride + c]); // global_atomic_add_f32
  }
  {
    const int g = tid & (kG - 1);
    if (tid < kG) {
      const unsigned cgrp = cnt[g];
      if (cgrp != 0u)
        atomicAdd(out + (size_t)kFeatElems + (size_t)b * kG + g, (float)cgrp);
    }
  }
}

__global__ __launch_bounds__(256) void finalize_kernel(float* __restrict__ out) {
  const int i = blockIdx.x * blockDim.x + threadIdx.x;   // over B*G
  if (i >= kMaskElems) return;
  const float s = 1.0f / (float)kN;
  float4* f4 = (float4*)(out + (size_t)i * kC);
  #pragma unroll
  for (int j = 0; j < kC / 4; ++j) {
    float4 v = f4[j];
    v.x *= s; v.y *= s; v.z *= s; v.w *= s;
    f4[j] = v;
  }
  float* maskp = out + (size_t)kFeatElems;
  maskp[i] = (maskp[i] > 0.f) ? 1.f : 0.f;
}

extern "C" void kernel_launch(void* const* d_in, const int* in_sizes, int n_in,
                              void* d_out, int out_size, void* d_ws, size_t ws_size,
                              hipStream_t stream) {
  (void)in_sizes; (void)n_in; (void)d_ws; (void)ws_size; (void)out_size;
  const float* x = (const float*)d_in[0];
  const int*   y = (const int*)d_in[1];
  float*     out = (float*)d_out;

  const int n4 = (kFeatElems + kMaskElems) / 4;  // 133120 float4
  zero_out_kernel<<<(n4 + 255) / 256, 256, 0, stream>>>((float4*)out, n4);

  dim3 grid(kChunks, kB);
  const size_t smem = (size_t)kG * kAccStride * sizeof(float)  // 139264
                    + (size_t)kG * sizeof(unsigned)            //   2048
                    + (size_t)kPts * sizeof(int);              //   2048 -> 143360 B (< 320 KB)
  scatter_kernel<<<grid, kThreads, smem, stream>>>(x, y, out);

  finalize_kernel<<<(kMaskElems + 255) / 256, 256, 0, stream>>>(out);
}